// SparseConvUNet_49400713838616
// MI455X (gfx1250) — compile-verified
//
#include <hip/hip_runtime.h>

// ---------------------------------------------------------------------------
// CDNA5 (gfx1250) fp32 sparse-UNet forward.
// All 3x3x3 and 2x2x2 convolutions run on V_WMMA_F32_16X16X4_F32 (wave32).
// Each wave computes MT=2 16-voxel tiles x all Cout/16 channel tiles with
// register-blocked fragment reuse. Compile-time <D,Cin,Cout> kills runtime
// division; neighbor handling is clamp+cndmask (no EXEC manipulation around
// WMMA). A-fragments load as b64. BN/ReLU, up-sample, 1x1 shortcuts and
// concat/add are plain VALU kernels.
// ---------------------------------------------------------------------------

typedef __attribute__((ext_vector_type(2))) float v2f;
typedef __attribute__((ext_vector_type(8))) float v8f;

#define BN_EPS 1e-4f

__device__ __forceinline__ v8f wmma4(v2f a, v2f b, v8f c) {
  // D = A(16x4) * B(4x16) + C(16x16), fp32 throughout.
  return __builtin_amdgcn_wmma_f32_16x16x4_f32(
      /*neg_a=*/false, a, /*neg_b=*/false, b,
      /*c_mod=*/(short)0, c, /*reuse_a=*/false, /*reuse_b=*/false);
}

__device__ __forceinline__ int clampi(int v, int lo, int hi) {
  return v < lo ? lo : (v > hi ? hi : v);   // lowers to v_med3_i32 / cndmask
}

// ---------------------------------------------------------------------------
// SubMConv3d k=3 pad=1, output masked. One wave per (MT*16 voxels, Cout) tile.
// Weights layout: (3,3,3,Cin,Cout) row-major  ->  w[(tap*Cin + c)*Cout + o].
// ---------------------------------------------------------------------------
template <int D, int Cin, int Cout, int MT>
__global__ __launch_bounds__(32) void k_subm3_t(
    const float* __restrict__ in, const float* __restrict__ w,
    const unsigned char* __restrict__ mask, float* __restrict__ out) {
  constexpr int NT = Cout / 16;
  const int lane = threadIdx.x;
  const int half = lane >> 4;   // K-half select for A/B fragments
  const int l16  = lane & 15;   // A-row / B-column within tile
  const int v0   = blockIdx.x * (16 * MT);

  // A-matrix row voxel per m-tile (both wave halves hold M = lane&15).
  int xm[MT], ym[MT], zm[MT];
#pragma unroll
  for (int mt = 0; mt < MT; ++mt) {
    const int vm = v0 + mt * 16 + l16;
    xm[mt] = vm % D;
    ym[mt] = (vm / D) % D;
    zm[mt] = vm / (D * D);
  }

  v8f acc[MT][NT];
  {
    v8f z = {};
#pragma unroll
    for (int mt = 0; mt < MT; ++mt)
#pragma unroll
      for (int nt = 0; nt < NT; ++nt) acc[mt][nt] = z;
  }

#pragma unroll 1
  for (int dz = -1; dz <= 1; ++dz) {
#pragma unroll 1
    for (int dy = -1; dy <= 1; ++dy) {
#pragma unroll 1
      for (int dx = -1; dx <= 1; ++dx) {
        const int tap = ((dz + 1) * 3 + (dy + 1)) * 3 + (dx + 1);
        const float* wbase = w + (tap * Cin) * Cout + l16;
        // speculative prefetch of next tap's weights (global_prefetch_b8)
        __builtin_prefetch(wbase + Cin * Cout, 0, 0);

        const float* arow[MT];
        bool inb[MT];
#pragma unroll
        for (int mt = 0; mt < MT; ++mt) {
          const int nx = xm[mt] + dx, ny = ym[mt] + dy, nz = zm[mt] + dz;
          inb[mt] = ((unsigned)nx < (unsigned)D) &
                    ((unsigned)ny < (unsigned)D) &
                    ((unsigned)nz < (unsigned)D);
          // clamp coords -> always-valid address, no pointer select/branch;
          // loaded value is select-zeroed below when out of bounds.
          const int nxc = clampi(nx, 0, D - 1);
          const int nyc = clampi(ny, 0, D - 1);
          const int nzc = clampi(nz, 0, D - 1);
          arow[mt] = in + ((nzc * D + nyc) * D + nxc) * Cin;
        }

#pragma unroll 4
        for (int kb = 0; kb < Cin; kb += 4) {
          const int ka = kb + half * 2;

          v2f bf[NT];
#pragma unroll
          for (int nt = 0; nt < NT; ++nt) {
            if constexpr ((Cin % 4) == 0) {
              v2f t = {wbase[ka * Cout + nt * 16],
                       wbase[(ka + 1) * Cout + nt * 16]};
              bf[nt] = t;
            } else {  // Cin==6 tail: clamp index, select-zero (no branches)
              const int k0 = (ka < Cin) ? ka : 0;
              const int k1 = (ka + 1 < Cin) ? ka + 1 : 0;
              const float b0 = wbase[k0 * Cout + nt * 16];
              const float b1 = wbase[k1 * Cout + nt * 16];
              v2f t = {(ka < Cin) ? b0 : 0.f, (ka + 1 < Cin) ? b1 : 0.f};
              bf[nt] = t;
            }
          }

          v2f af[MT];
#pragma unroll
          for (int mt = 0; mt < MT; ++mt) {
            if constexpr ((Cin % 4) == 0) {
              // 8B-aligned (rows are multiples of 8B, ka even): b64 load
              const v2f t = *(const v2f*)(arow[mt] + ka);
              v2f s = {inb[mt] ? t[0] : 0.f, inb[mt] ? t[1] : 0.f};
              af[mt] = s;
            } else {  // Cin==6: scalar clamped loads (avoid OOB b64 at tail)
              const int k0 = (ka < Cin) ? ka : 0;
              const int k1 = (ka + 1 < Cin) ? ka + 1 : 0;
              const float a0 = arow[mt][k0];
              const float a1 = arow[mt][k1];
              const bool p0 = inb[mt] && (ka < Cin);
              const bool p1 = inb[mt] && (ka + 1 < Cin);
              v2f s = {p0 ? a0 : 0.f, p1 ? a1 : 0.f};
              af[mt] = s;
            }
          }

#pragma unroll
          for (int mt = 0; mt < MT; ++mt)
#pragma unroll
            for (int nt = 0; nt < NT; ++nt)
              acc[mt][nt] = wmma4(af[mt], bf[nt], acc[mt][nt]);
        }
      }
    }
  }

  // C/D layout: VGPR r -> row (r + half*8), col = lane&15. Mask per voxel.
#pragma unroll
  for (int mt = 0; mt < MT; ++mt) {
#pragma unroll
    for (int r = 0; r < 8; ++r) {
      const int v = v0 + mt * 16 + r + half * 8;
      const bool act = mask[v] != 0;
#pragma unroll
      for (int nt = 0; nt < NT; ++nt)
        out[v * Cout + nt * 16 + l16] = act ? acc[mt][nt][r] : 0.f;
    }
  }
}

// ---------------------------------------------------------------------------
// SparseConv3d k=2 s=2 (down-sample). Output is naturally zero where the
// pooled mask is zero, so no explicit masking (matches reference).
// ---------------------------------------------------------------------------
template <int Dc, int Cin, int Cout, int MT>
__global__ __launch_bounds__(32) void k_down2_t(
    const float* __restrict__ in, const float* __restrict__ w,
    float* __restrict__ out) {
  constexpr int NT = Cout / 16;
  constexpr int Df = 2 * Dc;
  const int lane = threadIdx.x;
  const int half = lane >> 4;
  const int l16  = lane & 15;
  const int v0   = blockIdx.x * (16 * MT);

  int xc[MT], yc[MT], zc[MT];
#pragma unroll
  for (int mt = 0; mt < MT; ++mt) {
    const int vm = v0 + mt * 16 + l16;
    xc[mt] = vm % Dc;
    yc[mt] = (vm / Dc) % Dc;
    zc[mt] = vm / (Dc * Dc);
  }

  v8f acc[MT][NT];
  {
    v8f z = {};
#pragma unroll
    for (int mt = 0; mt < MT; ++mt)
#pragma unroll
      for (int nt = 0; nt < NT; ++nt) acc[mt][nt] = z;
  }

#pragma unroll 1
  for (int tap = 0; tap < 8; ++tap) {  // (i,j,k) with k fastest, all in-bounds
    const float* wbase = w + (tap * Cin) * Cout + l16;
    __builtin_prefetch(wbase + Cin * Cout, 0, 0);
    const float* arow[MT];
#pragma unroll
    for (int mt = 0; mt < MT; ++mt) {
      const int fx = 2 * xc[mt] + (tap & 1);
      const int fy = 2 * yc[mt] + ((tap >> 1) & 1);
      const int fz = 2 * zc[mt] + (tap >> 2);
      arow[mt] = in + ((fz * Df + fy) * Df + fx) * Cin;
    }
#pragma unroll 4
    for (int kb = 0; kb < Cin; kb += 4) {  // Cin in {32,64}: always aligned
      const int ka = kb + half * 2;
      v2f bf[NT];
#pragma unroll
      for (int nt = 0; nt < NT; ++nt) {
        v2f t = {wbase[ka * Cout + nt * 16], wbase[(ka + 1) * Cout + nt * 16]};
        bf[nt] = t;
      }
      v2f af[MT];
#pragma unroll
      for (int mt = 0; mt < MT; ++mt)
        af[mt] = *(const v2f*)(arow[mt] + ka);   // b64 load
#pragma unroll
      for (int mt = 0; mt < MT; ++mt)
#pragma unroll
        for (int nt = 0; nt < NT; ++nt)
          acc[mt][nt] = wmma4(af[mt], bf[nt], acc[mt][nt]);
    }
  }

#pragma unroll
  for (int mt = 0; mt < MT; ++mt)
#pragma unroll
    for (int r = 0; r < 8; ++r) {
      const int v = v0 + mt * 16 + r + half * 8;
#pragma unroll
      for (int nt = 0; nt < NT; ++nt)
        out[v * Cout + nt * 16 + l16] = acc[mt][nt][r];
    }
}

// ---------------------------------------------------------------------------
// SparseInverseConv3d k=2 (up-sample), masked to fine-level indices.
// out[q,o] = sum_c in[parent(q),c] * w[(i,j,k),c,o],  (i,j,k)=(qz&1,qy&1,qx&1)
// ---------------------------------------------------------------------------
__global__ void k_up2(const float* __restrict__ in, const float* __restrict__ w,
                      const unsigned char* __restrict__ mask,
                      float* __restrict__ out, int Dc, int Cin, int Cout) {
  const int Df = 2 * Dc;
  const int Nf = Df * Df * Df;
  const int idx = blockIdx.x * blockDim.x + threadIdx.x;
  if (idx >= Nf * Cout) return;
  const int o = idx % Cout;
  const int q = idx / Cout;
  if (!mask[q]) { out[idx] = 0.f; return; }
  const int qx = q % Df, qy = (q / Df) % Df, qz = q / (Df * Df);
  const int p = ((qz >> 1) * Dc + (qy >> 1)) * Dc + (qx >> 1);
  const int tap = (((qz & 1) * 2 + (qy & 1)) * 2) + (qx & 1);
  const float* arow = in + p * Cin;
  const float* wrow = w + (tap * Cin) * Cout + o;
  float s = 0.f;
  for (int c = 0; c < Cin; ++c) s += arow[c] * wrow[c * Cout];
  out[idx] = s;
}

// 1x1 shortcut: out[v,o] = sum_c in[v,c] * w[o,c]   (w shape (Cout,Cin))
__global__ void k_lin(const float* __restrict__ in, const float* __restrict__ w,
                      float* __restrict__ out, int N, int Cin, int Cout) {
  const int idx = blockIdx.x * blockDim.x + threadIdx.x;
  if (idx >= N * Cout) return;
  const int o = idx % Cout;
  const int v = idx / Cout;
  const float* arow = in + v * Cin;
  const float* wrow = w + o * Cin;
  float s = 0.f;
  for (int c = 0; c < Cin; ++c) s += arow[c] * wrow[c];
  out[idx] = s;
}

__global__ void k_concat(const float* __restrict__ a, const float* __restrict__ b,
                         float* __restrict__ out, int N, int Ca, int Cb) {
  const int C = Ca + Cb;
  const int idx = blockIdx.x * blockDim.x + threadIdx.x;
  if (idx >= N * C) return;
  const int c = idx % C;
  const int v = idx / C;
  out[idx] = (c < Ca) ? a[v * Ca + c] : b[v * Cb + (c - Ca)];
}

__global__ void k_add(const float* __restrict__ a, const float* __restrict__ b,
                      float* __restrict__ out, int n) {
  const int i = blockIdx.x * blockDim.x + threadIdx.x;
  if (i < n) out[i] = a[i] + b[i];
}

__global__ void k_zero(float* __restrict__ p, int n) {
  const int i = blockIdx.x * blockDim.x + threadIdx.x;
  if (i < n) p[i] = 0.f;
}

// mask pooling: any-of-8-children
__global__ void k_pool(const unsigned char* __restrict__ mf,
                       unsigned char* __restrict__ mc, int Dc) {
  const int i = blockIdx.x * blockDim.x + threadIdx.x;
  const int Nc = Dc * Dc * Dc;
  if (i >= Nc) return;
  const int x = i % Dc, y = (i / Dc) % Dc, z = i / (Dc * Dc);
  const int Df = 2 * Dc;
  unsigned char r = 0;
  for (int t = 0; t < 8; ++t) {
    const int fx = 2 * x + (t & 1);
    const int fy = 2 * y + ((t >> 1) & 1);
    const int fz = 2 * z + (t >> 2);
    r |= mf[(fz * Df + fy) * Df + fx];
  }
  mc[i] = r ? 1 : 0;
}

__global__ void k_count(const unsigned char* __restrict__ m, float* __restrict__ out, int N) {
  __shared__ float sh[256];
  const int tid = threadIdx.x;
  float s = 0.f;
  for (int i = blockIdx.x * blockDim.x + tid; i < N; i += gridDim.x * blockDim.x)
    s += m[i] ? 1.f : 0.f;
  sh[tid] = s;
  __syncthreads();
  for (int st = 128; st > 0; st >>= 1) {
    if (tid < st) sh[tid] += sh[tid + st];
    __syncthreads();
  }
  if (tid == 0) atomicAdd(out, sh[0]);
}

// per-channel sum / sumsq over active voxels. blockDim.x == C.
__global__ void k_bn_stats(const float* __restrict__ in,
                           const unsigned char* __restrict__ mask,
                           float* __restrict__ st, int N, int C) {
  const int c = threadIdx.x;
  float s = 0.f, ss = 0.f;
  for (int v = blockIdx.x; v < N; v += gridDim.x) {
    if (mask[v]) {
      const float t = in[v * C + c];
      s += t;
      ss += t * t;
    }
  }
  atomicAdd(&st[c], s);
  atomicAdd(&st[C + c], ss);
}

// out = relu((x - mean) * rsqrt(var + eps)) at active voxels, 0 elsewhere.
__global__ void k_bn_apply(const float* __restrict__ in,
                           const unsigned char* __restrict__ mask,
                           const float* __restrict__ st,
                           const float* __restrict__ cnt,
                           float* __restrict__ out, int N, int C) {
  const int idx = blockIdx.x * blockDim.x + threadIdx.x;
  if (idx >= N * C) return;
  const int c = idx % C;
  const int v = idx / C;
  const float n = cnt[0];
  const float mean = st[c] / n;
  const float var  = st[C + c] / n - mean * mean;  // == E[(x-mean)^2 * m]
  float val = 0.f;
  if (mask[v]) val = (in[idx] - mean) * rsqrtf(var + BN_EPS);
  out[idx] = fmaxf(val, 0.f);
}

// ---------------------------------------------------------------------------
extern "C" void kernel_launch(void* const* d_in, const int* in_sizes, int n_in,
                              void* d_out, int out_size, void* d_ws, size_t ws_size,
                              hipStream_t stream) {
  (void)in_sizes; (void)n_in; (void)out_size; (void)ws_size;

  const float*         x     = (const float*)d_in[0];
  const unsigned char* M1    = (const unsigned char*)d_in[1];  // bool mask
  const float* w_in  = (const float*)d_in[2];
  const float* w_b1a = (const float*)d_in[3];
  const float* w_b1b = (const float*)d_in[4];
  const float* w_d1  = (const float*)d_in[5];
  const float* w_b2a = (const float*)d_in[6];
  const float* w_b2b = (const float*)d_in[7];
  const float* w_d2  = (const float*)d_in[8];
  const float* w_b3a = (const float*)d_in[9];
  const float* w_b3b = (const float*)d_in[10];
  const float* w_u2  = (const float*)d_in[11];
  const float* w_t2sc= (const float*)d_in[12];
  const float* w_t2a = (const float*)d_in[13];
  const float* w_t2b = (const float*)d_in[14];
  const float* w_u1  = (const float*)d_in[15];
  const float* w_t1sc= (const float*)d_in[16];
  const float* w_t1a = (const float*)d_in[17];
  const float* w_t1b = (const float*)d_in[18];
  float* OUT = (float*)d_out;

  const int D1 = 48, D2 = 24, D3 = 12;
  const int N1 = D1 * D1 * D1;   // 110592
  const int N2 = D2 * D2 * D2;   // 13824
  const int N3 = D3 * D3 * D3;   // 1728

  // bump allocator over workspace (~115 MB total)
  char* base = (char*)d_ws;
  size_t off = 0;
  auto alloc = [&](size_t bytes) -> void* {
    off = (off + 255) & ~(size_t)255;
    void* p = base + off;
    off += bytes;
    return p;
  };
  unsigned char* M2 = (unsigned char*)alloc(N2);
  unsigned char* M3 = (unsigned char*)alloc(N3);
  float* CNT  = (float*)alloc(4 * sizeof(float));       // n1,n2,n3
  float* ST   = (float*)alloc(256 * sizeof(float));     // sum[C], sumsq[C]
  float* X0   = (float*)alloc((size_t)N1 * 32 * 4);
  float* X1   = (float*)alloc((size_t)N1 * 32 * 4);
  float* H1   = (float*)alloc((size_t)N1 * 32 * 4);
  float* H2   = (float*)alloc((size_t)N1 * 32 * 4);
  float* CAT1 = (float*)alloc((size_t)N1 * 64 * 4);
  float* Y0   = (float*)alloc((size_t)N2 * 64 * 4);
  float* Y1   = (float*)alloc((size_t)N2 * 64 * 4);
  float* G1   = (float*)alloc((size_t)N2 * 64 * 4);
  float* G2   = (float*)alloc((size_t)N2 * 64 * 4);
  float* U2   = (float*)alloc((size_t)N2 * 64 * 4);
  float* CAT2 = (float*)alloc((size_t)N2 * 128 * 4);
  float* Z0   = (float*)alloc((size_t)N3 * 96 * 4);
  float* Z1   = (float*)alloc((size_t)N3 * 96 * 4);
  float* K1   = (float*)alloc((size_t)N3 * 96 * 4);
  float* K2   = (float*)alloc((size_t)N3 * 96 * 4);

  auto bn = [&](const float* in, const unsigned char* m, const float* cnt,
                float* out_, int N, int C) {
    k_zero<<<dim3(1), dim3(256), 0, stream>>>(ST, 2 * C);
    k_bn_stats<<<dim3(128), dim3(C), 0, stream>>>(in, m, ST, N, C);
    const int tot = N * C;
    k_bn_apply<<<dim3((tot + 255) / 256), dim3(256), 0, stream>>>(in, m, ST, cnt, out_, N, C);
  };
  auto add = [&](const float* a, const float* b, float* o, int n) {
    k_add<<<dim3((n + 255) / 256), dim3(256), 0, stream>>>(a, b, o, n);
  };

  // ---- masks and active-voxel counts ----
  k_pool<<<dim3((N2 + 255) / 256), dim3(256), 0, stream>>>(M1, M2, D2);
  k_pool<<<dim3((N3 + 255) / 256), dim3(256), 0, stream>>>(M2, M3, D3);
  k_zero<<<dim3(1), dim3(256), 0, stream>>>(CNT, 4);
  k_count<<<dim3(64), dim3(256), 0, stream>>>(M1, CNT + 0, N1);
  k_count<<<dim3(64), dim3(256), 0, stream>>>(M2, CNT + 1, N2);
  k_count<<<dim3(64), dim3(256), 0, stream>>>(M3, CNT + 2, N3);

  const dim3 W32(32);
  const dim3 G1d(N1 / 32), G2d(N2 / 32), G3d(N3 / 32);  // MT=2 -> 32 voxels/wave

  // ---- encoder ----
  k_subm3_t<48, 6, 32, 2><<<G1d, W32, 0, stream>>>(x, w_in, M1, X0);      // input_conv
  bn(X0, M1, CNT + 0, H1, N1, 32);                                        // resblock1
  k_subm3_t<48, 32, 32, 2><<<G1d, W32, 0, stream>>>(H1, w_b1a, M1, H2);
  bn(H2, M1, CNT + 0, H1, N1, 32);
  k_subm3_t<48, 32, 32, 2><<<G1d, W32, 0, stream>>>(H1, w_b1b, M1, H2);
  add(H2, X0, X1, N1 * 32);                                               // x1
  bn(X1, M1, CNT + 0, H1, N1, 32);
  k_down2_t<24, 32, 64, 2><<<G2d, W32, 0, stream>>>(H1, w_d1, Y0);        // d1
  bn(Y0, M2, CNT + 1, G1, N2, 64);                                        // resblock2
  k_subm3_t<24, 64, 64, 2><<<G2d, W32, 0, stream>>>(G1, w_b2a, M2, G2);
  bn(G2, M2, CNT + 1, G1, N2, 64);
  k_subm3_t<24, 64, 64, 2><<<G2d, W32, 0, stream>>>(G1, w_b2b, M2, G2);
  add(G2, Y0, Y1, N2 * 64);                                               // x2
  bn(Y1, M2, CNT + 1, G1, N2, 64);
  k_down2_t<12, 64, 96, 2><<<G3d, W32, 0, stream>>>(G1, w_d2, Z0);        // d2
  bn(Z0, M3, CNT + 2, K1, N3, 96);                                        // resblock3
  k_subm3_t<12, 96, 96, 2><<<G3d, W32, 0, stream>>>(K1, w_b3a, M3, K2);
  bn(K2, M3, CNT + 2, K1, N3, 96);
  k_subm3_t<12, 96, 96, 2><<<G3d, W32, 0, stream>>>(K1, w_b3b, M3, K2);
  add(K2, Z0, Z1, N3 * 96);                                               // x3

  // ---- decoder level 2 ----
  bn(Z1, M3, CNT + 2, K1, N3, 96);
  k_up2<<<dim3((N2 * 64 + 255) / 256), dim3(256), 0, stream>>>(K1, w_u2, M2, U2, D3, 96, 64);
  k_concat<<<dim3((N2 * 128 + 255) / 256), dim3(256), 0, stream>>>(Y1, U2, CAT2, N2, 64, 64);
  k_lin<<<dim3((N2 * 64 + 255) / 256), dim3(256), 0, stream>>>(CAT2, w_t2sc, G1, N2, 128, 64); // shortcut
  bn(CAT2, M2, CNT + 1, CAT2, N2, 128);                                   // in-place (elementwise)
  k_subm3_t<24, 128, 64, 2><<<G2d, W32, 0, stream>>>(CAT2, w_t2a, M2, G2);
  bn(G2, M2, CNT + 1, Y0, N2, 64);
  k_subm3_t<24, 64, 64, 2><<<G2d, W32, 0, stream>>>(Y0, w_t2b, M2, G2);
  add(G2, G1, Y1, N2 * 64);                                               // t2

  // ---- decoder level 1 ----
  bn(Y1, M2, CNT + 1, G1, N2, 64);
  k_up2<<<dim3((N1 * 32 + 255) / 256), dim3(256), 0, stream>>>(G1, w_u1, M1, X0, D2, 64, 32); // u1 -> X0
  k_concat<<<dim3((N1 * 64 + 255) / 256), dim3(256), 0, stream>>>(X1, X0, CAT1, N1, 32, 32);
  k_lin<<<dim3((N1 * 32 + 255) / 256), dim3(256), 0, stream>>>(CAT1, w_t1sc, H1, N1, 64, 32); // shortcut
  bn(CAT1, M1, CNT + 0, CAT1, N1, 64);                                    // in-place
  k_subm3_t<48, 64, 32, 2><<<G1d, W32, 0, stream>>>(CAT1, w_t1a, M1, H2);
  bn(H2, M1, CNT + 0, X0, N1, 32);
  k_subm3_t<48, 32, 32, 2><<<G1d, W32, 0, stream>>>(X0, w_t1b, M1, H2);
  add(H2, H1, X1, N1 * 32);                                               // t1

  // ---- output layer: BN + ReLU -> d_out ----
  bn(X1, M1, CNT + 0, OUT, N1, 32);
}